// PaiNNInteraction_34686155882966
// MI455X (gfx1250) — compile-verified
//
#include <hip/hip_runtime.h>

// PaiNN interaction for MI455X (gfx1250, wave32).
// Phase 1: per-atom MLP via V_WMMA_F32_16X16X4_F32 (fp32 tensor path), x -> d_ws.
//          Requires ws_size >= N*192*4 bytes (15.4 MB for N=20000).
// Phase 2: sorted-segment gather/reduce, one wave per atom, no atomics.
//          Wij/dir streamed with non-temporal loads so x/mu tables stay in the
//          192MB L2; output written with NT stores.

typedef __attribute__((ext_vector_type(2))) float v2f;
typedef __attribute__((ext_vector_type(8))) float v8f;

#define FDIM 64
#define F3   192
#define HSTR 66   // LDS row stride (floats): even (8B-aligned float2), bank-spread

__device__ __forceinline__ v8f wmma_f32_4(v2f a, v2f b, v8f c) {
  // 8 args: (neg_a, A, neg_b, B, c_mod, C, reuse_a, reuse_b)
  return __builtin_amdgcn_wmma_f32_16x16x4_f32(false, a, false, b, (short)0, c,
                                               false, false);
}

__device__ __forceinline__ float silu(float z) {
  // z * v_rcp_f32(1+exp(-z)): one TRANS op instead of an IEEE divide chain
  return z * __builtin_amdgcn_rcpf(1.0f + __expf(-z));
}

// ---------------------------------------------------------------------------
// Phase 1: x[n,:] = silu(q[n,:] @ W1 + b1) @ W2 + b2
// One wave per 16-atom tile; 64 WMMAs (layer1) + 192 WMMAs (layer2) per tile.
// A-operand fragments hoisted into registers and reused across column tiles.
// ---------------------------------------------------------------------------
__global__ void __launch_bounds__(64)
painn_mlp_wmma(const float* __restrict__ q, const float* __restrict__ W1,
               const float* __restrict__ b1, const float* __restrict__ W2,
               const float* __restrict__ b2, float* __restrict__ xw, int N) {
  __shared__ float hbuf[2][16 * HSTR];

  const int lane = threadIdx.x & 31;
  const int wv   = threadIdx.x >> 5;          // wave in block: 0 or 1
  const int ntiles = (N + 15) >> 4;
  int tile = blockIdx.x * 2 + wv;
  if (tile >= ntiles) tile = ntiles - 1;      // clamp: keep EXEC all-ones

  const int half = lane >> 4;                 // 0: lanes 0-15, 1: lanes 16-31
  const int l16  = lane & 15;
  const int koff = half * 2;                  // K sub-offset per ISA layout
  const int row  = l16;                       // A-matrix row owned by lane

  int arow = tile * 16 + row;
  if (arow >= N) arow = N - 1;
  const float* qrow = q + (size_t)arow * FDIM;
  float* h = hbuf[wv];

  v2f aops[16];  // A fragments for this lane: reused across all column tiles

  // ---- layer 1: [16x64] = [16x64] @ [64x64], 4 column tiles ----
#pragma unroll
  for (int kk = 0; kk < 16; ++kk)
    aops[kk] = *(const v2f*)(qrow + kk * 4 + koff);

  for (int nt = 0; nt < 4; ++nt) {
    const int col = nt * 16 + l16;
    v8f c = {0.f, 0.f, 0.f, 0.f, 0.f, 0.f, 0.f, 0.f};
#pragma unroll
    for (int kk = 0; kk < 16; ++kk) {
      const int k = kk * 4 + koff;
      v2f b;
      b.x = W1[(size_t)k * FDIM + col];
      b.y = W1[(size_t)(k + 1) * FDIM + col];
      c = wmma_f32_4(aops[kk], b, c);
    }
    const float bias = b1[col];
#pragma unroll
    for (int r = 0; r < 8; ++r) {
      h[(half * 8 + r) * HSTR + col] = silu(c[r] + bias);
    }
  }
  __syncthreads();  // LDS store -> ds_load ordering (block always full)

  // ---- layer 2: [16x192] = [16x64] @ [64x192], 12 column tiles ----
#pragma unroll
  for (int kk = 0; kk < 16; ++kk)
    aops[kk] = *(const v2f*)(h + row * HSTR + kk * 4 + koff);

  for (int nt = 0; nt < 12; ++nt) {
    const int col = nt * 16 + l16;
    v8f c = {0.f, 0.f, 0.f, 0.f, 0.f, 0.f, 0.f, 0.f};
#pragma unroll
    for (int kk = 0; kk < 16; ++kk) {
      const int k = kk * 4 + koff;
      v2f b;
      b.x = W2[(size_t)k * F3 + col];
      b.y = W2[(size_t)(k + 1) * F3 + col];
      c = wmma_f32_4(aops[kk], b, c);
    }
    const float bias = b2[col];
#pragma unroll
    for (int r = 0; r < 8; ++r) {
      const int orow = tile * 16 + half * 8 + r;
      if (orow < N) xw[(size_t)orow * F3 + col] = c[r] + bias;
    }
  }
}

// ---------------------------------------------------------------------------
// Phase 2: one wave per atom i. Binary-search the sorted idx_i for this
// atom's pair range, accumulate dq/dmu in registers, write q+dq / mu+dmu.
// Lane owns features {2*lane, 2*lane+1} (float2, fully coalesced).
// ---------------------------------------------------------------------------
__global__ void __launch_bounds__(256)
painn_gather(const float* __restrict__ q, const float* __restrict__ mu,
             const float* __restrict__ Wij, const float* __restrict__ dir_ij,
             const int* __restrict__ idx_i, const int* __restrict__ idx_j,
             const float* __restrict__ x, float* __restrict__ out,
             int N, int P) {
  const int wave = (int)(blockIdx.x * (blockDim.x >> 5) + (threadIdx.x >> 5));
  const int lane = threadIdx.x & 31;
  if (wave >= N) return;
  const int a = wave;
  const int f = lane * 2;

  // lower_bound(idx_i, a) and lower_bound(idx_i, a+1)
  int lo = 0, hi = P;
  while (lo < hi) {
    int m = (lo + hi) >> 1;
    if (idx_i[m] < a) lo = m + 1; else hi = m;
  }
  const int p0 = lo;
  hi = P;
  while (lo < hi) {
    int m = (lo + hi) >> 1;
    if (idx_i[m] < a + 1) lo = m + 1; else hi = m;
  }
  const int p1 = lo;

  // accumulators start at q / mu (output = input + delta)
  v2f accq  = *(const v2f*)(q  + (size_t)a * FDIM + f);
  v2f accm0 = *(const v2f*)(mu + (size_t)a * F3 + 0 * FDIM + f);
  v2f accm1 = *(const v2f*)(mu + (size_t)a * F3 + 1 * FDIM + f);
  v2f accm2 = *(const v2f*)(mu + (size_t)a * F3 + 2 * FDIM + f);

  for (int p = p0; p < p1; ++p) {
    const int j = idx_j[p];
    const float* wp = Wij + (size_t)p * F3 + f;
    const float* xp = x   + (size_t)j * F3 + f;
    const float* mp = mu  + (size_t)j * F3 + f;

    // prefetch the Wij stream a few rows ahead (global_prefetch_b8)
    __builtin_prefetch(wp + 4 * F3, 0, 0);

    // 491MB one-shot stream: non-temporal so x/mu stay resident in L2
    v2f w0 = __builtin_nontemporal_load((const v2f*)(wp));
    v2f w1 = __builtin_nontemporal_load((const v2f*)(wp + FDIM));
    v2f w2 = __builtin_nontemporal_load((const v2f*)(wp + 2 * FDIM));
    // gathered tables (15MB each): regular temporal -> L2 hits
    v2f x0 = *(const v2f*)(xp);
    v2f x1 = *(const v2f*)(xp + FDIM);
    v2f x2 = *(const v2f*)(xp + 2 * FDIM);
    v2f m0 = *(const v2f*)(mp);
    v2f m1 = *(const v2f*)(mp + FDIM);
    v2f m2 = *(const v2f*)(mp + 2 * FDIM);
    float d0 = __builtin_nontemporal_load(dir_ij + (size_t)p * 3 + 0);
    float d1 = __builtin_nontemporal_load(dir_ij + (size_t)p * 3 + 1);
    float d2 = __builtin_nontemporal_load(dir_ij + (size_t)p * 3 + 2);

    accq += w0 * x0;            // dq
    v2f dR = w1 * x1;           // dmuR
    v2f dM = w2 * x2;           // dmumu
    accm0 += dR * d0 + dM * m0;
    accm1 += dR * d1 + dM * m1;
    accm2 += dR * d2 + dM * m2;
  }

  __builtin_nontemporal_store(accq, (v2f*)(out + (size_t)a * FDIM + f));
  float* om = out + (size_t)N * FDIM + (size_t)a * F3 + f;
  __builtin_nontemporal_store(accm0, (v2f*)(om));
  __builtin_nontemporal_store(accm1, (v2f*)(om + FDIM));
  __builtin_nontemporal_store(accm2, (v2f*)(om + 2 * FDIM));
}

extern "C" void kernel_launch(void* const* d_in, const int* in_sizes, int n_in,
                              void* d_out, int out_size, void* d_ws,
                              size_t ws_size, hipStream_t stream) {
  const float* q     = (const float*)d_in[0];
  const float* mu    = (const float*)d_in[1];
  const float* Wij   = (const float*)d_in[2];
  const float* dir   = (const float*)d_in[3];
  const int*   idx_i = (const int*)d_in[4];
  const int*   idx_j = (const int*)d_in[5];
  // d_in[6] = n_atoms scalar; derive from sizes instead
  const float* W1 = (const float*)d_in[7];
  const float* b1 = (const float*)d_in[8];
  const float* W2 = (const float*)d_in[9];
  const float* b2 = (const float*)d_in[10];

  const int N = in_sizes[0] / FDIM;  // q is [N,1,F]
  const int P = in_sizes[4];         // idx_i is [P]

  float* xw  = (float*)d_ws;  // [N,192] intermediate, 15.4MB for N=20000
  float* out = (float*)d_out; // [N,64] q_out ++ [N,3,64] mu_out

  const int ntiles = (N + 15) / 16;
  const int mlp_blocks = (ntiles + 1) / 2;  // 2 waves (tiles) per 64-thread block
  painn_mlp_wmma<<<mlp_blocks, 64, 0, stream>>>(q, W1, b1, W2, b2, xw, N);

  const int gblocks = (N * 32 + 255) / 256;  // one wave32 per atom
  painn_gather<<<gblocks, 256, 0, stream>>>(q, mu, Wij, dir, idx_i, idx_j, xw,
                                            out, N, P);
}